// DecoderRNN_55542517072229
// MI455X (gfx1250) — compile-verified
//
#include <hip/hip_runtime.h>
#include <hip/hip_bf16.h>

// ---------------------------------------------------------------------------
// DecoderRNN (image-caption LSTM) on MI455X / gfx1250, fp32-exact via
// V_WMMA_F32_16X16X4_F32.
//   B=64, T=32, E=H=512, V=10000
// ---------------------------------------------------------------------------

typedef float v2f __attribute__((ext_vector_type(2)));
typedef float v8f __attribute__((ext_vector_type(8)));

#define BB 64
#define TT 32
#define EE 512
#define HH 512
#define VV 10000
#define G4 2048   // 4*H

// LDS tile stride (floats): 32 cols + 4 pad -> 144B rows (16B aligned),
// row*36 mod 64 hits distinct 4k bank groups for 16 consecutive rows.
#define LDP 36

// ---------------------------------------------------------------------------
// Generic WMMA GEMM: C[M,N] = A[M,K](row) * B[N,K](row)^T  + epilogue
//   mode 0: out[r*N + c] = acc + bias0[c] + bias1[c]
//   mode 1: out[r*N + c] = acc + addend[r*N + c]
//   mode 2: out[((r%64)*TT + r/64)*N + c] = acc + bias0[c]   (guard c < N)
// Block tile: 64(M) x 128(N), K-chunk 32, 256 threads = 8 waves,
// each wave computes a 64x16 strip (4 accumulators of v8f).
// ---------------------------------------------------------------------------
__global__ __launch_bounds__(256)
void decrnn_gemm_wmma_f32(const float* __restrict__ A, int lda,
                          const float* __restrict__ Bm, int ldb,
                          float* __restrict__ out,
                          const float* __restrict__ bias0,
                          const float* __restrict__ bias1,
                          const float* __restrict__ addend,
                          int M, int N, int K, int mode)
{
    __shared__ float sA[64 * LDP];
    __shared__ float sB[128 * LDP];

    const int tid   = threadIdx.x;
    const int lane  = tid & 31;
    const int wave  = tid >> 5;          // 0..7 -> 16-wide N strip
    const int row16 = lane & 15;
    const int hi    = lane >> 4;         // 0 or 1

    const int m0 = blockIdx.x * 64;
    const int n0 = blockIdx.y * 128;

    v8f acc[4];
#pragma unroll
    for (int m = 0; m < 4; ++m) acc[m] = (v8f)0.0f;

    const int lr = tid >> 3;             // 0..31  (row within staging step)
    const int lc = (tid & 7) * 4;        // 0,4,...,28 (float4 column)

    for (int k0 = 0; k0 < K; k0 += 32) {
        // ---- stage A tile: 64 rows x 32 cols (2 float4 per thread) ----
#pragma unroll
        for (int i = 0; i < 2; ++i) {
            int r    = lr + i * 32;
            int grow = m0 + r;
            float4 v = make_float4(0.f, 0.f, 0.f, 0.f);
            if (grow < M)
                v = *(const float4*)(A + (size_t)grow * lda + k0 + lc);
            float* d = &sA[r * LDP + lc];
            d[0] = v.x; d[1] = v.y; d[2] = v.z; d[3] = v.w;
        }
        // ---- stage B tile: 128 rows x 32 cols (4 float4 per thread) ----
#pragma unroll
        for (int i = 0; i < 4; ++i) {
            int r    = lr + i * 32;
            int grow = n0 + r;
            float4 v = make_float4(0.f, 0.f, 0.f, 0.f);
            if (grow < N)
                v = *(const float4*)(Bm + (size_t)grow * ldb + k0 + lc);
            float* d = &sB[r * LDP + lc];
            d[0] = v.x; d[1] = v.y; d[2] = v.z; d[3] = v.w;
        }
        __syncthreads();

        // ---- compute: 8 x (K=4) WMMA steps on this chunk ----
#pragma unroll
        for (int kk = 0; kk < 32; kk += 4) {
            const int kcol = kk + 2 * hi;            // K = vgpr + 2*hi
            v2f bf;
            {
                const float* bp = &sB[(wave * 16 + row16) * LDP + kcol];
                bf.x = bp[0]; bf.y = bp[1];
            }
#pragma unroll
            for (int m = 0; m < 4; ++m) {
                v2f af;
                const float* ap = &sA[(m * 16 + row16) * LDP + kcol];
                af.x = ap[0]; af.y = ap[1];
                acc[m] = __builtin_amdgcn_wmma_f32_16x16x4_f32(
                    false, af, false, bf, (short)0, acc[m], false, false);
            }
        }
        __syncthreads();
    }

    // ---- epilogue ----
    const int col = n0 + wave * 16 + row16;          // N index for this lane
#pragma unroll
    for (int m = 0; m < 4; ++m) {
#pragma unroll
        for (int j = 0; j < 8; ++j) {
            int   grow = m0 + m * 16 + j + 8 * hi;   // M = vgpr + 8*hi
            float val  = acc[m][j];
            if (mode == 0) {
                out[(size_t)grow * N + col] = val + bias0[col] + bias1[col];
            } else if (mode == 1) {
                size_t idx = (size_t)grow * N + col;
                out[idx] = val + addend[idx];
            } else {
                if (col < N) {
                    int b = grow & 63;
                    int t = grow >> 6;
                    out[(size_t)(b * TT + t) * N + col] = val + bias0[col];
                }
            }
        }
    }
}

// ---------------------------------------------------------------------------
// Build X[T*B, E]: row r = t*64+b -> features (t==0) or embed_W[captions[b,t]]
// ---------------------------------------------------------------------------
__global__ __launch_bounds__(256)
void decrnn_gather_x(const float* __restrict__ feat,
                     const int* __restrict__ caps,
                     const float* __restrict__ embW,
                     float* __restrict__ X)
{
    int r = blockIdx.x;                  // 0..2047
    int t = r >> 6;
    int b = r & 63;
    const float* src = (t == 0) ? (feat + (size_t)b * EE)
                                : (embW + (size_t)caps[b * TT + t] * EE);
    float*       dst = X + (size_t)r * EE;
    ((float2*)dst)[threadIdx.x] = ((const float2*)src)[threadIdx.x]; // 256*2=512
}

// ---------------------------------------------------------------------------
// LSTM cell elementwise (gate order i,f,g,o):
//   c = sig(f)*c + sig(i)*tanh(g);  h = sig(o)*tanh(c)
// ---------------------------------------------------------------------------
__device__ __forceinline__ float dec_sig(float x) {
    return 1.0f / (1.0f + __expf(-x));
}

__global__ __launch_bounds__(256)
void decrnn_lstm_cell(const float* __restrict__ gates,
                      float* __restrict__ h,
                      float* __restrict__ c,
                      float* __restrict__ hseq_t)
{
    int idx = blockIdx.x * 256 + threadIdx.x;   // 0..32767
    int b   = idx >> 9;
    int n   = idx & 511;
    const float* g = gates + (size_t)b * G4;
    float ig = g[n];
    float fg = g[n + HH];
    float gg = g[n + 2 * HH];
    float og = g[n + 3 * HH];
    float cn = dec_sig(fg) * c[idx] + dec_sig(ig) * tanhf(gg);
    float hn = dec_sig(og) * tanhf(cn);
    c[idx]      = cn;
    h[idx]      = hn;
    hseq_t[idx] = hn;
}

// ---------------------------------------------------------------------------
// Host orchestration
// ---------------------------------------------------------------------------
extern "C" void kernel_launch(void* const* d_in, const int* in_sizes, int n_in,
                              void* d_out, int out_size, void* d_ws, size_t ws_size,
                              hipStream_t stream)
{
    (void)in_sizes; (void)n_in; (void)out_size; (void)ws_size;

    const float* features = (const float*)d_in[0];   // [64,512]
    const int*   captions = (const int*)  d_in[1];   // [64,32]
    const float* embed_W  = (const float*)d_in[2];   // [10000,512]
    const float* W_ih     = (const float*)d_in[3];   // [2048,512]
    const float* W_hh     = (const float*)d_in[4];   // [2048,512]
    const float* b_ih     = (const float*)d_in[5];   // [2048]
    const float* b_hh     = (const float*)d_in[6];   // [2048]
    const float* fc_W     = (const float*)d_in[7];   // [10000,512]
    const float* fc_b     = (const float*)d_in[8];   // [10000]
    float*       logits   = (float*)d_out;           // [64,32,10000]

    // workspace layout (floats)
    float* X      = (float*)d_ws;                    // [2048,512]
    float* x_proj = X      + (size_t)2048 * 512;     // [2048,2048]
    float* h_seq  = x_proj + (size_t)2048 * 2048;    // [2048,512]
    float* gates  = h_seq  + (size_t)2048 * 512;     // [64,2048]
    float* h      = gates  + (size_t)64 * 2048;      // [64,512]
    float* c      = h      + (size_t)64 * 512;       // [64,512]

    // zero initial recurrent state (capture-safe)
    hipMemsetAsync(h, 0, (size_t)64 * 512 * sizeof(float), stream);
    hipMemsetAsync(c, 0, (size_t)64 * 512 * sizeof(float), stream);

    // 1) build per-step inputs X
    decrnn_gather_x<<<2048, 256, 0, stream>>>(features, captions, embed_W, X);

    // 2) x_proj = X * W_ih^T + b_ih + b_hh      [2048 x 2048]
    decrnn_gemm_wmma_f32<<<dim3(32, 16), 256, 0, stream>>>(
        X, EE, W_ih, EE, x_proj, b_ih, b_hh, nullptr,
        2048, G4, EE, /*mode=*/0);

    // 3) sequential LSTM scan over T steps
    for (int t = 0; t < TT; ++t) {
        decrnn_gemm_wmma_f32<<<dim3(1, 16), 256, 0, stream>>>(
            h, HH, W_hh, HH, gates, nullptr, nullptr,
            x_proj + (size_t)t * 64 * G4,
            64, G4, HH, /*mode=*/1);
        decrnn_lstm_cell<<<128, 256, 0, stream>>>(
            gates, h, c, h_seq + (size_t)t * 64 * HH);
    }

    // 4) logits[b,t,:] = h_seq[t*64+b,:] * fc_W^T + fc_b   (permuted store)
    decrnn_gemm_wmma_f32<<<dim3(32, (VV + 127) / 128), 256, 0, stream>>>(
        h_seq, HH, fc_W, HH, logits, fc_b, nullptr, nullptr,
        2048, VV, HH, /*mode=*/2);
}